// TwoLayerGNN_54855322304849
// MI455X (gfx1250) — compile-verified
//
#include <hip/hip_runtime.h>

// CDNA5 / gfx1250, wave32.
typedef __attribute__((ext_vector_type(2))) float v2f;
typedef __attribute__((ext_vector_type(8))) float v8f;

#define HSTRIDE 132   // 128 + 4 dword pad: A-fragment reads (16 rows, same k) hit distinct LDS banks

// ---------------- zero scratch ----------------
__global__ void zero_kernel(float* __restrict__ p, long n) {
  long i = (long)blockIdx.x * blockDim.x + threadIdx.x;
  long stride = (long)gridDim.x * blockDim.x;
  for (; i < n; i += stride) p[i] = 0.0f;
}

// ---------------- layer-1 edge aggregation (6-dim, raw x) ----------------
__global__ void agg1_kernel(const float* __restrict__ x,
                            const long long* __restrict__ ei,  // [2][E] int64
                            float* __restrict__ sum1,          // [N][6]
                            float* __restrict__ deg,           // [N]
                            long E) {
  long e = (long)blockIdx.x * blockDim.x + threadIdx.x;
  if (e >= E) return;
  long src = ei[e];
  long dst = ei[E + e];
  atomicAdd(&deg[dst], 1.0f);
#pragma unroll
  for (int c = 0; c < 6; ++c)
    atomicAdd(&sum1[dst * 6 + c], x[src * 6 + c]);
}

// ---------------- fused: layer1 (VALU, K=12) + layer2 GEMM (WMMA f32 16x16x4) ----------------
// Block: 256 threads = 8 waves, handles 64 nodes.
//   h[64][128] = relu([mean1 | x] @ [W1_l ; W1_r] + b1)   (in LDS)
//   pr[64][128] = h @ [W2_l | W2_r]                        (WMMA, to global)
__global__ __launch_bounds__(256)
void l1gemm_kernel(const float* __restrict__ x,     // [N][6]
                   const float* __restrict__ sum1,  // [N][6]
                   const float* __restrict__ deg,   // [N]
                   const float* __restrict__ W1l,   // [6][128]
                   const float* __restrict__ b1,    // [128]
                   const float* __restrict__ W1r,   // [6][128]
                   const float* __restrict__ W2l,   // [128][64]
                   const float* __restrict__ W2r,   // [128][64]
                   float* __restrict__ pr,          // [N][128]
                   int N) {
  __shared__ float sW1[12 * 128];      // rows 0..5 = W1_l, rows 6..11 = W1_r
  __shared__ float sB1[128];
  __shared__ float sMX[64 * 12];       // per node: mean1[0..5], x[6..11]
  __shared__ float sH[64 * HSTRIDE];   // h tile, padded row stride

  const int tid  = threadIdx.x;
  const int base = blockIdx.x * 64;

  for (int i = tid; i < 6 * 128; i += 256) sW1[i] = W1l[i];
  for (int i = tid; i < 6 * 128; i += 256) sW1[6 * 128 + i] = W1r[i];
  if (tid < 128) sB1[tid] = b1[tid];

  if (tid < 64) {
    int g = base + tid;
    float d = (g < N) ? deg[g] : 1.0f;
    float inv = 1.0f / fmaxf(d, 1.0f);
#pragma unroll
    for (int k = 0; k < 6; ++k) {
      float mv = 0.0f, xv = 0.0f;
      if (g < N) {
        mv = sum1[(long)g * 6 + k] * inv;
        xv = x[(long)g * 6 + k];
      }
      sMX[tid * 12 + k]     = mv;
      sMX[tid * 12 + 6 + k] = xv;
    }
  }
  __syncthreads();

  // h tile: 64*128 elements / 256 threads = 32 per thread; ln uniform per wave.
#pragma unroll 4
  for (int it = 0; it < 32; ++it) {
    int idx = it * 256 + tid;
    int ln  = idx >> 7;
    int col = idx & 127;
    float acc = sB1[col];
#pragma unroll
    for (int k = 0; k < 6; ++k) {
      acc += sMX[ln * 12 + k]     * sW1[k * 128 + col];
      acc += sMX[ln * 12 + 6 + k] * sW1[(6 + k) * 128 + col];
    }
    sH[ln * HSTRIDE + col] = fmaxf(acc, 0.0f);
  }
  __syncthreads();

  // ---- WMMA: each wave owns a 16-row tile x one 64-col half; 4 col tiles x 32 k-steps.
  const int wave = tid >> 5;
  const int lane = tid & 31;
  const int rt   = wave & 3;                 // row tile 0..3 (16 rows each)
  const int ch   = wave >> 2;                // column half: 0 -> W2_l, 1 -> W2_r
  const float* __restrict__ WB = ch ? W2r : W2l;   // [128][64]
  const int m    = lane & 15;                // A/B fragment row/col within tile
  const int koff = (lane >> 4) * 2;          // upper half-wave holds K+2,K+3

  // Hoist all 32 A fragments for this wave's row tile (64 VGPRs).
  v2f aF[32];
#pragma unroll
  for (int ks = 0; ks < 32; ++ks) {
    int k = ks * 4 + koff;                   // even k -> 8B aligned LDS read
    aF[ks] = *(const v2f*)&sH[(rt * 16 + m) * HSTRIDE + k];
  }

  for (int ct = 0; ct < 4; ++ct) {
    v8f acc = {};
    const int colLocal = ct * 16 + m;        // column within 64-wide half (L1/L2-resident weight reads)
#pragma unroll
    for (int ks = 0; ks < 32; ++ks) {
      int k = ks * 4 + koff;
      v2f bF;
      bF.x = WB[k * 64 + colLocal];
      bF.y = WB[(k + 1) * 64 + colLocal];
      acc = __builtin_amdgcn_wmma_f32_16x16x4_f32(
          /*neg_a=*/false, aF[ks], /*neg_b=*/false, bF,
          /*c_mod=*/(short)0, acc, /*reuse_a=*/false, /*reuse_b=*/false);
    }
    // C/D layout: VGPR r, lanes 0-15 -> M=r, lanes 16-31 -> M=r+8; N = lane&15
    const int colg = ch * 64 + ct * 16 + (lane & 15);
#pragma unroll
    for (int r = 0; r < 8; ++r) {
      int row = rt * 16 + r + ((lane >> 4) << 3);
      int g = base + row;
      if (g < N) pr[(long)g * 128 + colg] = acc[r];
    }
  }
}

// ---------------- layer-2 edge aggregation: wave per edge, 64-dim p rows ----------------
__global__ void agg2_kernel(const float* __restrict__ pr,    // [N][128], cols 0..63 = p
                            const long long* __restrict__ ei,
                            float* __restrict__ agg2,        // [N][64]
                            long E) {
  long e = ((long)blockIdx.x * blockDim.x + threadIdx.x) >> 5;
  int lane = threadIdx.x & 31;
  if (e >= E) return;
  long src = ei[e];
  long dst = ei[E + e];
  float p0 = pr[src * 128 + lane];
  float p1 = pr[src * 128 + lane + 32];
  atomicAdd(&agg2[dst * 64 + lane], p0);
  atomicAdd(&agg2[dst * 64 + lane + 32], p1);
}

// ---------------- epilogue ----------------
__global__ void final_kernel(const float* __restrict__ agg2,
                             const float* __restrict__ deg,
                             const float* __restrict__ b2,   // [64]
                             const float* __restrict__ pr,   // cols 64..127 = h@W2_r
                             float* __restrict__ out,        // [N][64]
                             int N) {
  long idx = (long)blockIdx.x * blockDim.x + threadIdx.x;
  if (idx >= (long)N * 64) return;
  long i = idx >> 6;
  int  c = idx & 63;
  float inv = 1.0f / fmaxf(deg[i], 1.0f);
  out[idx] = agg2[idx] * inv + b2[c] + pr[i * 128 + 64 + c];
}

extern "C" void kernel_launch(void* const* d_in, const int* in_sizes, int n_in,
                              void* d_out, int out_size, void* d_ws, size_t ws_size,
                              hipStream_t stream) {
  const float*     x   = (const float*)d_in[0];
  const long long* ei  = (const long long*)d_in[1];
  const float*     W1l = (const float*)d_in[2];
  const float*     b1  = (const float*)d_in[3];
  const float*     W1r = (const float*)d_in[4];
  const float*     W2l = (const float*)d_in[5];
  const float*     b2  = (const float*)d_in[6];
  const float*     W2r = (const float*)d_in[7];
  float* out = (float*)d_out;

  const int  N = in_sizes[0] / 6;
  const long E = (long)in_sizes[1] / 2;

  // Workspace layout (all f32): pr[N*128] | agg2[N*64] | sum1[N*6] | deg[N]  (~80 MB)
  float* pr   = (float*)d_ws;
  float* agg2 = pr   + (size_t)N * 128;
  float* sum1 = agg2 + (size_t)N * 64;
  float* degp = sum1 + (size_t)N * 6;

  // agg2 / sum1 / deg are contiguous -> one zero pass
  long zn = (long)N * 64 + (long)N * 6 + N;
  zero_kernel<<<1024, 256, 0, stream>>>(agg2, zn);

  agg1_kernel<<<(int)((E + 255) / 256), 256, 0, stream>>>(x, ei, sum1, degp, E);

  l1gemm_kernel<<<(N + 63) / 64, 256, 0, stream>>>(x, sum1, degp,
                                                   W1l, b1, W1r, W2l, W2r, pr, N);

  agg2_kernel<<<(int)((E + 7) / 8), 256, 0, stream>>>(pr, ei, agg2, E);

  final_kernel<<<(int)(((long)N * 64 + 255) / 256), 256, 0, stream>>>(
      agg2, degp, b2, pr, out, N);
}